// SmallSMBlock_55585466744989
// MI455X (gfx1250) — compile-verified
//
#include <hip/hip_runtime.h>
#include <stdint.h>

// ---------------- problem constants ----------------
#define Hd 512
#define Wd 512
#define NIMG 32
#define TILE 16           // 16x16 pixel tile per block (stage 2)
#define IPB  8            // images per block (stage 2)
#define KPS  12           // Kp row stride in floats (9 used + 3 pad -> 48B, b128-friendly)

typedef float v2f __attribute__((ext_vector_type(2)));
typedef float v8f __attribute__((ext_vector_type(8)));
typedef int   i32x4 __attribute__((vector_size(16)));   // matches builtin's expected pointee

// ---------------- CDNA5 async global->LDS helpers ----------------
__device__ __forceinline__ void async_copy_b32(void* lds_dst, const void* g_src) {
#if __has_builtin(__builtin_amdgcn_global_load_async_to_lds_b32)
  __builtin_amdgcn_global_load_async_to_lds_b32(
      (__attribute__((address_space(1))) int*)g_src,
      (__attribute__((address_space(3))) int*)lds_dst, 0, 0);
#else
  uint32_t loff = (uint32_t)(uintptr_t)(__attribute__((address_space(3))) void*)lds_dst;
  asm volatile("global_load_async_to_lds_b32 %0, %1, off"
               :: "v"(loff), "v"(g_src) : "memory");
#endif
}

__device__ __forceinline__ void async_copy_b128(void* lds_dst, const void* g_src) {
#if __has_builtin(__builtin_amdgcn_global_load_async_to_lds_b128)
  __builtin_amdgcn_global_load_async_to_lds_b128(
      (__attribute__((address_space(1))) i32x4*)g_src,
      (__attribute__((address_space(3))) i32x4*)lds_dst, 0, 0);
#else
  uint32_t loff = (uint32_t)(uintptr_t)(__attribute__((address_space(3))) void*)lds_dst;
  asm volatile("global_load_async_to_lds_b128 %0, %1, off"
               :: "v"(loff), "v"(g_src) : "memory");
#endif
}

__device__ __forceinline__ void wait_async0() {
#if __has_builtin(__builtin_amdgcn_s_wait_asynccnt)
  __builtin_amdgcn_s_wait_asynccnt(0);
#else
  asm volatile("s_wait_asynccnt 0" ::: "memory");
#endif
}

// ---------------- stage 1: per-pixel kernel field via f32 WMMA ----------------
// K[p, c] = sum_t patch(image, p, t) * KLw[c, t]  + KLb[c]
// GEMM per wave: M=16 pixels, N=16 (9 channels used), contraction k=9 -> 3x k=4 WMMA.
__device__ __forceinline__ float tapval(const float* __restrict__ img, int h, int w, int t) {
  if (t >= 9) return 0.0f;
  int i = t / 3, j = t - 3 * i;
  int hh = h + i - 1, ww = w + j - 1;
  return ((unsigned)hh < (unsigned)Hd && (unsigned)ww < (unsigned)Wd) ? img[hh * Wd + ww] : 0.0f;
}

__device__ __forceinline__ float wval(const float* __restrict__ KLw, int c, int t) {
  return (c < 9 && t < 9) ? KLw[c * 9 + t] : 0.0f;
}

__global__ __launch_bounds__(256) void kfield_wmma(const float* __restrict__ image,
                                                   const float* __restrict__ KLw,
                                                   const float* __restrict__ KLb,
                                                   float* __restrict__ Kp) {
  const int lane = threadIdx.x & 31;
  const int wv   = threadIdx.x >> 5;
  const int p0   = (blockIdx.x * 8 + wv) * 16;   // 16 pixels per wave
  const int m    = lane & 15;                    // A-matrix row / C column index
  const int hi   = lane >> 4;                    // lane half selects K pair {0,1} vs {2,3}
  const int p    = p0 + m;
  const int h    = p >> 9, w = p & (Wd - 1);

  v8f acc = {0.f, 0.f, 0.f, 0.f, 0.f, 0.f, 0.f, 0.f};
#pragma unroll
  for (int t0 = 0; t0 < 12; t0 += 4) {
    const int ka = t0 + hi * 2;
    v2f a, b;
    // A (16x4 f32): lanes 0-15 hold K={t0,t0+1} for M=lane; lanes 16-31 hold K={t0+2,t0+3}
    a.x = tapval(image, h, w, ka);
    a.y = tapval(image, h, w, ka + 1);
    // B (4x16 f32): column N = lane&15 (output channel), rows = same K pairing
    b.x = wval(KLw, m, ka);
    b.y = wval(KLw, m, ka + 1);
    acc = __builtin_amdgcn_wmma_f32_16x16x4_f32(false, a, false, b, (short)0, acc,
                                                false, false);
  }

  // D layout: VGPR r -> M = r + 8*hi, N = lane&15. Write channel c = m for 8 pixels.
  if (m < 9) {
    const float bias = KLb[m];
#pragma unroll
    for (int r = 0; r < 8; ++r) {
      const int mm = r + hi * 8;
      Kp[(size_t)(p0 + mm) * KPS + m] = acc[r] + bias;
    }
  }
}

// ---------------- stage 2: apply spatially-varying 3x3 filter ----------------
// Bandwidth-bound: stage x halo tile + Kp tile into LDS via async loads,
// reuse each x element 9x and each Kp vector across IPB images.
__global__ __launch_bounds__(256) void kapply(const float* __restrict__ x,
                                              const float* __restrict__ Kp,
                                              float* __restrict__ y) {
  __shared__ float xs[IPB][TILE + 2][TILE + 4];   // row stride 20 -> conflict-free
  __shared__ float kt[TILE * TILE][KPS];          // 48B per pixel, b128-aligned

  const int tid = threadIdx.x;
  const int tx = tid & 15, ty = tid >> 4;
  const int w0 = blockIdx.x * TILE;
  const int h0 = blockIdx.y * TILE;
  const int imgBase = blockIdx.z * IPB;

  // Kp tile: one pixel per thread, 3 x b128 async copies (48B)
  {
    const int p = (h0 + ty) * Wd + (w0 + tx);
    const float* src = Kp + (size_t)p * KPS;
#pragma unroll
    for (int q = 0; q < 3; ++q) async_copy_b128(&kt[tid][q * 4], src + q * 4);
  }

  // x halo tiles: 18x18 per image, async b32 per element; zero-fill OOB halo
#pragma unroll 1
  for (int im = 0; im < IPB; ++im) {
    const float* xim = x + (size_t)(imgBase + im) * (Hd * Wd);
    for (int e = tid; e < 18 * 18; e += 256) {
      const int r = e / 18, c = e - r * 18;
      const int gh = h0 - 1 + r, gw = w0 - 1 + c;
      if ((unsigned)gh < (unsigned)Hd && (unsigned)gw < (unsigned)Wd)
        async_copy_b32(&xs[im][r][c], xim + gh * Wd + gw);
      else
        xs[im][r][c] = 0.0f;
    }
  }

  wait_async0();
  __syncthreads();

  float k[9];
#pragma unroll
  for (int t = 0; t < 9; ++t) k[t] = kt[tid][t];

#pragma unroll
  for (int im = 0; im < IPB; ++im) {
    float acc = 0.0f;
#pragma unroll
    for (int i = 0; i < 3; ++i)
#pragma unroll
      for (int j = 0; j < 3; ++j)
        acc = fmaf(xs[im][ty + i][tx + j], k[3 * i + j], acc);
    y[(size_t)(imgBase + im) * (Hd * Wd) + (size_t)(h0 + ty) * Wd + (w0 + tx)] = acc;
  }
}

// ---------------- host launcher ----------------
extern "C" void kernel_launch(void* const* d_in, const int* in_sizes, int n_in,
                              void* d_out, int out_size, void* d_ws, size_t ws_size,
                              hipStream_t stream) {
  const float* image = (const float*)d_in[0];   // (1, 512, 512)
  const float* x     = (const float*)d_in[1];   // (32, 1, 512, 512)
  const float* KLw   = (const float*)d_in[2];   // (9, 1, 3, 3)
  const float* KLb   = (const float*)d_in[3];   // (9,)
  float*       y     = (float*)d_out;           // (32, 512, 512)
  float*       Kp    = (float*)d_ws;            // (512*512, 12) padded kernel field

  // stage 1: 512*512 pixels / (8 waves * 16 pixels) per block
  kfield_wmma<<<(Hd * Wd) / 128, 256, 0, stream>>>(image, KLw, KLb, Kp);

  // stage 2: 16x16 pixel tiles, 32/IPB image groups
  dim3 grid(Wd / TILE, Hd / TILE, NIMG / IPB);
  kapply<<<grid, 256, 0, stream>>>(x, Kp, y);
}